// Logic_Model_38302518346183
// MI455X (gfx1250) — compile-verified
//
#include <hip/hip_runtime.h>

// Problem constants (from reference)
#define SS 4096
#define HH 3
#define EE 512
#define GG 3334
#define RES 0.03f
#define ROWS (SS * HH)   // 12288 blocks, one per (s,h) row

typedef __attribute__((ext_vector_type(2))) float v2f;
typedef __attribute__((ext_vector_type(8))) float v8f;

// Sum a float across all 32 lanes of the wave using the CDNA5 matrix pipe:
// A = 16x4 matrix holding the 32 per-lane partials (VGPR1 = 0),
// B = 4x16 all-ones  ->  D[m,n] = rowsum(m) = acc[m] + acc[m+16].
// Per-lane sum of the 8 D VGPRs gives half-wave totals; one shfl_xor(16)
// completes the full-wave sum (valid in every lane).
__device__ __forceinline__ float wave_sum32(float x) {
  v2f a; a[0] = x;    a[1] = 0.0f;
  v2f b; b[0] = 1.0f; b[1] = 1.0f;
  v8f c = {};
  v8f d = __builtin_amdgcn_wmma_f32_16x16x4_f32(
      /*neg_a=*/false, a, /*neg_b=*/false, b,
      /*c_mod=*/(short)0, c, /*reuse_a=*/false, /*reuse_b=*/false);
  float s = ((d[0] + d[1]) + (d[2] + d[3])) + ((d[4] + d[5]) + (d[6] + d[7]));
  s += __shfl_xor(s, 16, 32);
  return s;
}

// Block-wide (256 threads = 8 waves) sum; result valid in thread 0.
__device__ __forceinline__ float block_sum256(float x) {
  __shared__ float lds[8];
  float ws = wave_sum32(x);
  const int lane = threadIdx.x & 31;
  const int wv   = threadIdx.x >> 5;
  if (lane == 0) lds[wv] = ws;
  __syncthreads();
  float t = 0.0f;
  if (threadIdx.x == 0) {
#pragma unroll
    for (int i = 0; i < 8; ++i) t += lds[i];
  }
  return t;
}

// One block per (s,h) row: streams G grid points (exp path) and E events
// (masked log path), emits   partial[row] = log_part - 0.03 * exp_part.
__global__ void __launch_bounds__(256)
logic_model_row_kernel(const float* __restrict__ evf,
                       const unsigned char* __restrict__ mask,
                       const float* __restrict__ grf,
                       const float* __restrict__ wts,
                       const float* __restrict__ bss,
                       const float* __restrict__ eff,
                       float* __restrict__ partial) {
  const int row = blockIdx.x;          // row = s*H + h
  const int h   = row % HH;
  const int tid = threadIdx.x;

  const float w    = wts[h] * eff[h];  // F == 1
  const float base = bss[h];

  float accExp = 0.0f;   // sum exp(base + w*x) over grid points
  float accLog = 0.0f;   // sum (base + w*x) over masked events

  // ---- grid/integral part: rows alternate 16B/8B alignment -> peel + vec4 ----
  const size_t goff = (size_t)row * GG;
  const float* grow = grf + goff;
  const int pre = (int)((4u - ((unsigned)goff & 3u)) & 3u);   // 0 or 2
  for (int g = tid; g < pre; g += 256)
    accExp += __expf(fmaf(w, grow[g], base));

  const int nv = (GG - pre) >> 2;                             // 833 exactly
  const float4* gv = (const float4*)(grow + pre);
  for (int i = tid; i < nv; i += 256) {
    const float4 v = gv[i];
    accExp += __expf(fmaf(w, v.x, base));
    accExp += __expf(fmaf(w, v.y, base));
    accExp += __expf(fmaf(w, v.z, base));
    accExp += __expf(fmaf(w, v.w, base));
  }
  for (int g = pre + (nv << 2) + tid; g < GG; g += 256)       // 0 or 2 tail
    accExp += __expf(fmaf(w, grow[g], base));

  // ---- event part: rows are float4/uchar4 aligned (E = 512) ----
  const size_t eoff = (size_t)row * EE;
  const float4* ev4 = (const float4*)(evf + eoff);
  const uchar4* mk4 = (const uchar4*)(mask + eoff);
  for (int i = tid; i < (EE >> 2); i += 256) {
    const float4 v = ev4[i];
    const uchar4 m = mk4[i];
    accLog += m.x ? fmaf(w, v.x, base) : 0.0f;
    accLog += m.y ? fmaf(w, v.y, base) : 0.0f;
    accLog += m.z ? fmaf(w, v.z, base) : 0.0f;
    accLog += m.w ? fmaf(w, v.w, base) : 0.0f;
  }

  const float t = block_sum256(accLog - RES * accExp);
  if (threadIdx.x == 0) partial[row] = t;
}

// Deterministic final reduction of the 12288 per-row partials.
__global__ void __launch_bounds__(256)
logic_model_final_kernel(const float* __restrict__ partial, int n,
                         float* __restrict__ out) {
  float acc = 0.0f;
  for (int i = threadIdx.x; i < n; i += 256) acc += partial[i];
  const float t = block_sum256(acc);
  if (threadIdx.x == 0) out[0] = t;
}

extern "C" void kernel_launch(void* const* d_in, const int* in_sizes, int n_in,
                              void* d_out, int out_size, void* d_ws, size_t ws_size,
                              hipStream_t stream) {
  // setup_inputs() order:
  //   0: event_features [S,H,E,1] f32
  //   1: event_mask     [S,H,E]   bool (1 byte/elem)
  //   2: grid_features  [S,H,G,1] f32
  //   3: weights        [H,1]     f32
  //   4: bases          [H]       f32
  //   5: effects        [H,1]     f32
  const float*         evf  = (const float*)d_in[0];
  const unsigned char* mask = (const unsigned char*)d_in[1];
  const float*         grf  = (const float*)d_in[2];
  const float*         wts  = (const float*)d_in[3];
  const float*         bss  = (const float*)d_in[4];
  const float*         eff  = (const float*)d_in[5];

  float* partial = (float*)d_ws;   // ROWS floats = 48 KB of scratch
  float* out     = (float*)d_out;

  logic_model_row_kernel<<<ROWS, 256, 0, stream>>>(evf, mask, grf, wts, bss, eff,
                                                   partial);
  logic_model_final_kernel<<<1, 256, 0, stream>>>(partial, ROWS, out);
}